// FrameDropout_37254546325873
// MI455X (gfx1250) — compile-verified
//
#include <hip/hip_runtime.h>

// Frame dropout: out[b,d,t] = x[b,d,t] * (u[t] >= 0.2f ? 1.0f : 0.0f)
// x: [32, 512, 4096] fp32 (t contiguous), u: [4096] fp32.
// Memory-bound: 512 MB of traffic -> ~22 us at 23.3 TB/s. No matrix math, so
// WMMA is inapplicable; optimize for b128 vector memory ops + NT streaming.

typedef __attribute__((ext_vector_type(4))) float v4f;

#define DROPOUT_PROB 0.2f

__global__ __launch_bounds__(256) void FrameDropout_kernel_v4(
    const v4f* __restrict__ x4,   // N/4 float4s, streaming (NT)
    const v4f* __restrict__ u4,   // T/4 float4s, L2-resident (16 KB total)
    v4f*       __restrict__ o4,   // N/4 float4s, streaming (NT)
    unsigned int n4,              // total float4 count
    unsigned int t4mask)          // (T/4 - 1), T/4 is a power of two
{
    unsigned int i = blockIdx.x * blockDim.x + threadIdx.x;
    if (i >= n4) return;

    // Mask vector for this thread's 4 consecutive t positions.
    // t index of element 4*i is (4*i) % T  ->  float4 index (i & t4mask).
    v4f u = u4[i & t4mask];                      // cached load (RT hint)

    // Streaming 128-bit NT load of x.
    v4f x = __builtin_nontemporal_load(&x4[i]);

    v4f m;
    m.x = (u.x >= DROPOUT_PROB) ? 1.0f : 0.0f;
    m.y = (u.y >= DROPOUT_PROB) ? 1.0f : 0.0f;
    m.z = (u.z >= DROPOUT_PROB) ? 1.0f : 0.0f;
    m.w = (u.w >= DROPOUT_PROB) ? 1.0f : 0.0f;

    // Multiply (matches reference semantics incl. NaN/Inf propagation),
    // then streaming 128-bit NT store.
    __builtin_nontemporal_store(x * m, &o4[i]);
}

// Fallback scalar kernel for shapes not divisible by 4 or with non-pow2 T/4
// (not hit by this harness's 32x512x4096 / 4096 shapes, but keeps the entry
// point deterministic and total).
__global__ __launch_bounds__(256) void FrameDropout_kernel_scalar(
    const float* __restrict__ x,
    const float* __restrict__ u,
    float*       __restrict__ o,
    unsigned int n,
    unsigned int T)
{
    unsigned int i = blockIdx.x * blockDim.x + threadIdx.x;
    if (i >= n) return;
    float m = (u[i % T] >= DROPOUT_PROB) ? 1.0f : 0.0f;
    o[i] = x[i] * m;
}

static inline bool is_pow2_u32(unsigned int v) { return v && ((v & (v - 1)) == 0); }

extern "C" void kernel_launch(void* const* d_in, const int* in_sizes, int n_in,
                              void* d_out, int out_size, void* d_ws, size_t ws_size,
                              hipStream_t stream) {
    (void)n_in; (void)out_size; (void)d_ws; (void)ws_size;

    const float* x = (const float*)d_in[0];   // [B, D, T] flat
    const float* u = (const float*)d_in[1];   // [T]
    float* out = (float*)d_out;

    const unsigned int n = (unsigned int)in_sizes[0];   // 32*512*4096
    const unsigned int T = (unsigned int)in_sizes[1];   // 4096

    const unsigned int threads = 256;

    if ((n % 4u) == 0u && (T % 4u) == 0u && is_pow2_u32(T / 4u)) {
        const unsigned int n4 = n / 4u;
        const unsigned int t4mask = (T / 4u) - 1u;
        const unsigned int blocks = (n4 + threads - 1u) / threads;
        FrameDropout_kernel_v4<<<blocks, threads, 0, stream>>>(
            (const v4f*)x, (const v4f*)u, (v4f*)out, n4, t4mask);
    } else {
        const unsigned int blocks = (n + threads - 1u) / threads;
        FrameDropout_kernel_scalar<<<blocks, threads, 0, stream>>>(
            x, u, out, n, T);
    }
}